// Conv_42872363548868
// MI455X (gfx1250) — compile-verified
//
#include <hip/hip_runtime.h>
#include <hip/hip_bf16.h>
#include <math.h>

// ---------------------------------------------------------------------------
// GNN conv layer: pre-FFN GEMM+GELU, edge gather*bases scatter-add, residual,
// FFN (2x GEMM+GELU), residual.  N=50000, E=800000, H=96.
// GEMMs use V_WMMA_F32_16X16X4_F32 (f32 in / f32 acc == reference numerics).
// Edge phase: NT stream of bases (307 MB read-once), hardware
// global_atomic_add_f32 at device scope into an L2-resident accumulator.
// ---------------------------------------------------------------------------

#define N_NODES 50000
#define N_EDGES 800000
#define HID     96
#define NODE_ELEMS (N_NODES * HID)   // 4,800,000 floats

typedef __attribute__((ext_vector_type(2))) float v2f;
typedef __attribute__((ext_vector_type(8))) float v8f;

__device__ __forceinline__ float gelu_erf(float x) {
    // exact (erf-based) GELU, matches jax.nn.gelu(approximate=False)
    return 0.5f * x * (1.0f + erff(x * 0.70710678118654752440f));
}

// Guaranteed-hardware f32 scatter atomic (no CAS expansion), device scope so
// adds from all WGPs combine coherently in L2 (CDNA5 SCOPE=DEV, ISA 7.1).
__device__ __forceinline__ void atomic_add_f32_hw(float* p, float v) {
    asm volatile("global_atomic_add_f32 %0, %1, off scope:SCOPE_DEV"
                 :: "v"(p), "v"(v)
                 : "memory");
}

// ---------------------------------------------------------------------------
// out[tileM*16 .. +16][tileN*16 .. +16] = act( A[rows][96] @ W[96][96] + b )
// optionally + resid.  One 16x16 tile per wave32, K swept in steps of 4 via
// v_wmma_f32_16x16x4_f32.
//
// Fragment layouts (ISA 7.12.2, f32):
//   A 16x4 : lane L -> row M=L&15 ; VGPR0 = A[M][2*(L>>4)], VGPR1 = A[M][2*(L>>4)+1]
//   B 4x16 : lane L -> col N=L&15 ; VGPR0 = B[2*(L>>4)][N], VGPR1 = B[2*(L>>4)+1][N]
//   C/D    : lane L -> col N=L&15 ; VGPR i = row (i + 8*(L>>4))
// ---------------------------------------------------------------------------
template<bool RESID>
__global__ __launch_bounds__(128)
void gemm96_gelu(const float* __restrict__ A,
                 const float* __restrict__ W,
                 const float* __restrict__ bias,
                 const float* __restrict__ resid,
                 float* __restrict__ out,
                 int nrows) {
    const int lane    = threadIdx.x & 31;
    const int wave    = threadIdx.x >> 5;
    const int tile    = blockIdx.x * 4 + wave;
    const int nTilesM = nrows >> 4;          // 3125
    const int tileM   = tile / 6;            // 6 column tiles of 16 over HID=96
    const int tileN   = tile % 6;
    if (tileM >= nTilesM) return;            // wave-uniform exit

    const int half   = lane >> 4;            // 0 | 1  -> K sub-pair selector
    const int lane16 = lane & 15;
    const int arow   = tileM * 16 + lane16;  // this lane's A row
    const int col    = tileN * 16 + lane16;  // this lane's output column

    // accumulator = bias broadcast down each column
    const float b = bias[col];
    v8f acc;
#pragma unroll
    for (int i = 0; i < 8; ++i) acc[i] = b;

    const float* ap = A + arow * HID + half * 2;        // + k0 each step
    const float* wp = W + half * 2 * HID + col;         // + k0*HID each step

#pragma unroll
    for (int k0 = 0; k0 < HID; k0 += 4) {
        v2f afrag = *(const v2f*)(ap + k0);             // A[M][k0+2h], A[M][k0+2h+1]
        v2f bfrag;
        bfrag.x = wp[k0 * HID];                         // W[k0+2h  ][N]
        bfrag.y = wp[k0 * HID + HID];                   // W[k0+2h+1][N]
        acc = __builtin_amdgcn_wmma_f32_16x16x4_f32(
            /*neg_a=*/false, afrag, /*neg_b=*/false, bfrag,
            /*c_mod=*/(short)0, acc, /*reuse_a=*/false, /*reuse_b=*/false);
    }

    const int rbase = tileM * 16 + half * 8;
#pragma unroll
    for (int i = 0; i < 8; ++i) {
        const int r = rbase + i;
        float v = gelu_erf(acc[i]);
        if (RESID) v += resid[r * HID + col];
        out[r * HID + col] = v;
    }
}

// ---------------------------------------------------------------------------
// accum starts as x_feat; add h[src]*bases into accum[dst].  One wave per
// edge, 3 features per lane (96 = 3*32), coalesced within the edge row.
// h (19.2 MB) and accum (19.2 MB) are L2-resident on MI455X (192 MB L2);
// bases is a 307 MB read-once stream -> non-temporal so it doesn't evict them.
// ---------------------------------------------------------------------------
__global__ __launch_bounds__(256)
void edge_scatter(const float* __restrict__ h,
                  const float* __restrict__ bases,
                  const int*   __restrict__ src,
                  const int*   __restrict__ dst,
                  float*       __restrict__ accum) {
    const int e    = (int)((blockIdx.x * 256u + threadIdx.x) >> 5);
    const int lane = threadIdx.x & 31;
    if (e >= N_EDGES) return;                // wave-uniform exit

    const int s = src[e];
    const int d = dst[e];
    const float* hp  = h     + (size_t)s * HID;
    const float* bp  = bases + (size_t)e * HID;
    float*       apd = accum + (size_t)d * HID;

#pragma unroll
    for (int j = 0; j < 3; ++j) {
        const int c = lane + 32 * j;
        const float bv = __builtin_nontemporal_load(bp + c);  // th:NT stream
        const float v  = hp[c] * bv;                          // L2-resident gather
        atomic_add_f32_hw(apd + c, v);                        // hw f32 atomic
    }
}

__global__ __launch_bounds__(256)
void copy_f4(const float4* __restrict__ in, float4* __restrict__ out, int n4) {
    const int i = blockIdx.x * 256 + threadIdx.x;
    if (i < n4) out[i] = in[i];
}

// ---------------------------------------------------------------------------
extern "C" void kernel_launch(void* const* d_in, const int* in_sizes, int n_in,
                              void* d_out, int out_size, void* d_ws, size_t ws_size,
                              hipStream_t stream) {
    const float* x_feat = (const float*)d_in[0];
    const float* bases  = (const float*)d_in[1];
    const float* W_pre  = (const float*)d_in[2];
    const float* b_pre  = (const float*)d_in[3];
    const float* W1     = (const float*)d_in[4];
    const float* b1     = (const float*)d_in[5];
    const float* W2     = (const float*)d_in[6];
    const float* b2     = (const float*)d_in[7];
    const int*   src    = (const int*)d_in[8];
    const int*   dst    = (const int*)d_in[9];
    float*       out    = (float*)d_out;

    float* ws_x = (float*)d_ws;                  // x_feat + aggr (residual 1)
    float* ws_h = ws_x + NODE_ELEMS;             // gelu(x @ W_pre + b_pre)
    float* ws_t = ws_h + NODE_ELEMS;             // gelu(x @ W1 + b1)

    const int n4 = NODE_ELEMS / 4;               // 1,200,000 float4
    const int nTiles = (N_NODES / 16) * 6;       // 18,750 wave-tiles
    const int gemmBlocks = (nTiles + 3) / 4;     // 4 waves / block
    const int edgeBlocks = N_EDGES / 8;          // 8 edge-waves / block

    // 1) seed accumulator with x_feat (residual-1 folded into scatter target)
    copy_f4<<<(n4 + 255) / 256, 256, 0, stream>>>(
        (const float4*)x_feat, (float4*)ws_x, n4);

    // 2) h = gelu(x_feat @ W_pre + b_pre)
    gemm96_gelu<false><<<gemmBlocks, 128, 0, stream>>>(
        x_feat, W_pre, b_pre, nullptr, ws_h, N_NODES);

    // 3) ws_x += scatter-add over edges of h[src] * bases
    edge_scatter<<<edgeBlocks, 256, 0, stream>>>(ws_h, bases, src, dst, ws_x);

    // 4) t = gelu(ws_x @ W1 + b1)
    gemm96_gelu<false><<<gemmBlocks, 128, 0, stream>>>(
        ws_x, W1, b1, nullptr, ws_t, N_NODES);

    // 5) out = ws_x + gelu(t @ W2 + b2)
    gemm96_gelu<true><<<gemmBlocks, 128, 0, stream>>>(
        ws_t, W2, b2, ws_x, out, N_NODES);
}